// HeteroModel_30983894073365
// MI455X (gfx1250) — compile-verified
//
#include <hip/hip_runtime.h>
#include <math.h>

typedef __attribute__((ext_vector_type(2))) float v2f;
typedef __attribute__((ext_vector_type(8))) float v8f;
typedef float __attribute__((address_space(1))) gfloat;

namespace {
constexpr int kN = 50000;   // nodes
constexpr int kE = 800000;  // edges per relation
constexpr int kR = 3;       // relations
constexpr int kD = 96;      // D_IN == D_HID (input width of both layers)
}

// force native global_atomic_add_f32 (agent scope, relaxed)
__device__ __forceinline__ void atomic_add_f32(float* p, float v) {
  __hip_atomic_fetch_add((gfloat*)(unsigned long long)p, v,
                         __ATOMIC_RELAXED, __HIP_MEMORY_SCOPE_AGENT);
}

// ---------------------------------------------------------------------------
// zero a region (used for deg + neigh scratch)
// ---------------------------------------------------------------------------
__global__ __launch_bounds__(256) void zero_f4(float4* __restrict__ p, long long n4) {
  long long i = (long long)blockIdx.x * blockDim.x + threadIdx.x;
  if (i < n4) p[i] = float4{0.f, 0.f, 0.f, 0.f};
}

// ---------------------------------------------------------------------------
// per-relation in-degree (integer atomics), then inv_deg = 1/max(deg,1)
// ---------------------------------------------------------------------------
__global__ __launch_bounds__(256) void degree_kernel(const int* __restrict__ dst,
                                                     int* __restrict__ deg) {
  int gid = blockIdx.x * 256 + threadIdx.x;  // exact grid: R*E threads
  int r = gid / kE;
  int d = dst[gid];
  atomicAdd(&deg[r * kN + d], 1);
}

__global__ __launch_bounds__(256) void invdeg_kernel(const int* __restrict__ deg,
                                                     float* __restrict__ inv) {
  int i = blockIdx.x * 256 + threadIdx.x;
  if (i < kR * kN) inv[i] = 1.0f / fmaxf((float)deg[i], 1.0f);
}

// ---------------------------------------------------------------------------
// edge scatter: neigh[r][d][:] += h[s][:] * inv_deg[r][d]
// 24 threads per edge (4 floats each), 8 edges per 192-thread block.
// ---------------------------------------------------------------------------
__global__ __launch_bounds__(192) void scatter_kernel(const float* __restrict__ h,
                                                      const int* __restrict__ src,
                                                      const int* __restrict__ dst,
                                                      const float* __restrict__ inv,
                                                      float* __restrict__ neigh) {
  int t = threadIdx.x;
  int e = blockIdx.x * 8 + t / 24;  // flattened (r, e); exact grid: R*E/8 blocks
  int c = t % 24;                   // 4-float chunk within the 96-wide feature
  int r = e / kE;
  int s = src[e];
  int d = dst[e];
  float w = inv[r * kN + d];
  const float4 v = *(const float4*)(h + (size_t)s * kD + c * 4);
  float* base = neigh + ((size_t)r * kN + d) * kD + c * 4;
  atomic_add_f32(base + 0, v.x * w);
  atomic_add_f32(base + 1, v.y * w);
  atomic_add_f32(base + 2, v.z * w);
  atomic_add_f32(base + 3, v.w * w);
}

// ---------------------------------------------------------------------------
// Fused SAGE layer: out = mean_r tanh([h|neigh_r] @ [Ws_r;Wn_r] + b_r)
// 32 rows x HOUT per workgroup (256 thr = 8 wave32). K = 192.
// A tile staged in LDS (row-major, stride 196); stacked weights staged in LDS
// TRANSPOSED to K-major (sW[n][k], stride 196) so both WMMA operands are
// 8B-aligned ds_load_b64 with conflict-free bank access.
//
// V_WMMA_F32_16X16X4_F32 layouts (ISA 7.12.2):
//   A 16x4 : lane<16 -> row=lane,    {K,K+1}; lane>=16 -> row=lane-16, {K+2,K+3}
//   B 4x16 : lane<16 -> col=lane,    {K,K+1}; lane>=16 -> col=lane-16, {K+2,K+3}
//   C/D    : vgpr i -> M = i + (lane>=16 ? 8 : 0), N = lane & 15
// ---------------------------------------------------------------------------
template <int HOUT>
__global__ __launch_bounds__(256) void rgcn_layer(const float* __restrict__ hin,    // N x 96
                                                  const float* __restrict__ neigh,  // R x N x 96 (pre-scaled)
                                                  const float* __restrict__ Ws,     // R x 96 x HOUT
                                                  const float* __restrict__ Wn,     // R x 96 x HOUT
                                                  const float* __restrict__ bias,   // R x HOUT
                                                  float* __restrict__ out,          // N x HOUT
                                                  int N) {
  constexpr int K2   = 2 * kD;    // 192 stacked [h | neigh]
  constexpr int LDA  = K2 + 4;    // 196 floats; 196 mod 64 = 4 -> conflict-free
  constexpr int LDB  = K2 + 4;    // transposed weight stride (K-major)
  constexpr int TM   = 32;        // rows per workgroup
  constexpr int NCT  = HOUT / 16; // column tiles (6 or 4)
  constexpr int JOBS = 2 * NCT;   // (rowtile, coltile) pairs (12 or 8)

  __shared__ float sA[TM * LDA];      // 32 x 196  (25.1 KB)
  __shared__ float sW[HOUT * LDB];    // HOUT x 196 (75.3 / 50.2 KB)

  const int m0    = blockIdx.x * TM;
  const int tid   = threadIdx.x;
  const int wave  = tid >> 5;
  const int lane  = tid & 31;
  const int lrow  = lane & 15;        // A-row / B-col / C-col selector
  const int khalf = (lane >> 4) * 2;  // K sub-offset per half-wave
  const int half8 = (lane >> 4) << 3; // C-row offset per half-wave

  v8f accm[2];
  accm[0] = v8f{0.f, 0.f, 0.f, 0.f, 0.f, 0.f, 0.f, 0.f};
  accm[1] = v8f{0.f, 0.f, 0.f, 0.f, 0.f, 0.f, 0.f, 0.f};

  // stage self-features once: columns [0,96). Tail block clamps row index.
  for (int idx = tid; idx < TM * 24; idx += 256) {
    int rr = idx / 24, cc = idx % 24;
    int g = m0 + rr; g = g < N ? g : N - 1;
    *(float4*)&sA[rr * LDA + cc * 4] = *(const float4*)&hin[(size_t)g * kD + cc * 4];
  }

  for (int r = 0; r < kR; ++r) {
    __syncthreads();  // previous-iteration readers done (also orders hin staging)

    // stage neighbor features for relation r: columns [96,192)
    for (int idx = tid; idx < TM * 24; idx += 256) {
      int rr = idx / 24, cc = idx % 24;
      int g = m0 + rr; g = g < N ? g : N - 1;
      *(float4*)&sA[rr * LDA + kD + cc * 4] =
          *(const float4*)&neigh[((size_t)r * N + g) * kD + cc * 4];
    }
    // stage stacked weights transposed: sW[n][k] = Ws_r[k][n] (k<96) / Wn_r[k-96][n]
    {
      const float* WsR = Ws + (size_t)r * kD * HOUT;
      const float* WnR = Wn + (size_t)r * kD * HOUT;
      constexpr int Q = kD * HOUT / 4;  // float4 count per half
      for (int idx = tid; idx < Q; idx += 256) {
        int k = idx / (HOUT / 4), n4 = idx % (HOUT / 4);
        float4 v = *(const float4*)&WsR[(size_t)k * HOUT + n4 * 4];
        sW[(n4 * 4 + 0) * LDB + k] = v.x;
        sW[(n4 * 4 + 1) * LDB + k] = v.y;
        sW[(n4 * 4 + 2) * LDB + k] = v.z;
        sW[(n4 * 4 + 3) * LDB + k] = v.w;
      }
      for (int idx = tid; idx < Q; idx += 256) {
        int k = idx / (HOUT / 4), n4 = idx % (HOUT / 4);
        float4 v = *(const float4*)&WnR[(size_t)k * HOUT + n4 * 4];
        sW[(n4 * 4 + 0) * LDB + kD + k] = v.x;
        sW[(n4 * 4 + 1) * LDB + kD + k] = v.y;
        sW[(n4 * 4 + 2) * LDB + kD + k] = v.z;
        sW[(n4 * 4 + 3) * LDB + kD + k] = v.w;
      }
    }
    __syncthreads();

    int ji = 0;
    for (int job = wave; job < JOBS; job += 8, ++ji) {
      const int rt = job & 1;        // row tile within the 32-row block
      const int n0 = (job >> 1) * 16;
      const float* aRow = &sA[(rt * 16 + lrow) * LDA];
      const float* bRow = &sW[(n0 + lrow) * LDB];

      const float bv = bias[r * HOUT + n0 + lrow];  // bias depends only on N
      v8f c = v8f{bv, bv, bv, bv, bv, bv, bv, bv};

#pragma unroll 8
      for (int k = 0; k < K2; k += 4) {
        v2f a = *(const v2f*)&aRow[k + khalf];  // ds_load_b64
        v2f b = *(const v2f*)&bRow[k + khalf];  // ds_load_b64
        c = __builtin_amdgcn_wmma_f32_16x16x4_f32(false, a, false, b,
                                                  (short)0, c, false, false);
      }
#pragma unroll
      for (int i = 0; i < 8; ++i) accm[ji][i] += tanhf(c[i]);
    }
  }

  // write mean over relations; lanes 0-15 each cover a 64B row segment
  const float inv3 = 1.0f / 3.0f;
  int ji = 0;
  for (int job = wave; job < JOBS; job += 8, ++ji) {
    const int rt  = job & 1;
    const int col = (job >> 1) * 16 + lrow;
    const int rb  = m0 + rt * 16 + half8;
#pragma unroll
    for (int i = 0; i < 8; ++i) {
      const int row = rb + i;
      if (row < N) out[(size_t)row * HOUT + col] = accm[ji][i] * inv3;
    }
  }
}

// ---------------------------------------------------------------------------
extern "C" void kernel_launch(void* const* d_in, const int* in_sizes, int n_in,
                              void* d_out, int out_size, void* d_ws, size_t ws_size,
                              hipStream_t stream) {
  const float* x   = (const float*)d_in[0];
  const int*   src = (const int*)d_in[1];
  const int*   dst = (const int*)d_in[2];
  const float* Ws1 = (const float*)d_in[3];
  const float* Wn1 = (const float*)d_in[4];
  const float* b1  = (const float*)d_in[5];
  const float* Ws2 = (const float*)d_in[6];
  const float* Wn2 = (const float*)d_in[7];
  const float* b2  = (const float*)d_in[8];
  float* out = (float*)d_out;

  // workspace layout: [deg:int RN][neigh:f32 RN*96][h1:f32 N*96][inv:f32 RN]
  const size_t RN = (size_t)kR * kN;
  char*  ws    = (char*)d_ws;
  int*   deg   = (int*)ws;
  float* neigh = (float*)(ws + RN * 4);
  float* h1    = (float*)(ws + RN * 4 + RN * (size_t)kD * 4);
  float* inv   = (float*)(ws + RN * 4 + RN * (size_t)kD * 4 + (size_t)kN * kD * 4);

  const int layerBlocks = (kN + 31) / 32;  // 1563 (last block is a 16-row tail)

  // 1) zero deg + neigh (contiguous: RN*(1+96) floats, divisible by 4)
  long long z4 = (long long)(RN * 97 / 4);
  zero_f4<<<(int)((z4 + 255) / 256), 256, 0, stream>>>((float4*)ws, z4);
  // 2) per-relation in-degree
  degree_kernel<<<(kR * kE) / 256, 256, 0, stream>>>(dst, deg);
  // 3) inverse degree
  invdeg_kernel<<<(kR * kN + 255) / 256, 256, 0, stream>>>(deg, inv);
  // 4) layer-1 aggregation: neigh = mean-scatter of x
  scatter_kernel<<<(kR * kE) / 8, 192, 0, stream>>>(x, src, dst, inv, neigh);
  // 5) layer 1 GEMM+tanh+mean -> h1
  rgcn_layer<96><<<layerBlocks, 256, 0, stream>>>(x, neigh, Ws1, Wn1, b1, h1, kN);
  // 6) re-zero neigh
  long long z4b = (long long)(RN * (size_t)kD / 4);
  zero_f4<<<(int)((z4b + 255) / 256), 256, 0, stream>>>((float4*)neigh, z4b);
  // 7) layer-2 aggregation on h1
  scatter_kernel<<<(kR * kE) / 8, 192, 0, stream>>>(h1, src, dst, inv, neigh);
  // 8) layer 2 GEMM+tanh+mean -> out
  rgcn_layer<64><<<layerBlocks, 256, 0, stream>>>(h1, neigh, Ws2, Wn2, b2, out, kN);
}